// SpeakerVectorLoss_62027917688920
// MI455X (gfx1250) — compile-verified
//
#include <hip/hip_runtime.h>
#include <hip/hip_bf16.h>
#include <math.h>

typedef __attribute__((ext_vector_type(2))) float v2f;
typedef __attribute__((ext_vector_type(4))) float v4f;
typedef __attribute__((ext_vector_type(8))) float v8f;

#define SPK   2000
#define DIM   256
#define TT    500
#define BB    8
#define NN    2
#define NCOL  (BB * NN * TT)      // 8000 columns total
#define NSTILE (SPK / 16)         // 125 s-tiles
#define LOG2E 1.44269504088896f
#define LN2   0.69314718055995f

// Raw hardware transcendentals (args always <= 0 / finite here, no range fixup needed)
#define FEXP2(x) __builtin_amdgcn_exp2f(x)
#define FLOG2(x) __builtin_amdgcn_logf(x)

__device__ __forceinline__ float wave_reduce_add(float v) {
  #pragma unroll
  for (int off = 16; off > 0; off >>= 1) v += __shfl_xor(v, off, 32);
  return v;
}

// Cross-half (lane ^ 16) swap as a single VALU op: v_permlanex16_b32, identity selects.
__device__ __forceinline__ float swap16(float v) {
  const int i = __float_as_int(v);
  return __int_as_float(
      __builtin_amdgcn_permlanex16(i, i, 0x76543210, 0xfedcba98, false, false));
}

// ---------------------------------------------------------------------------
// Kernel 1: E_S[:, :, :, 0] head output + zero the loss accumulator slot.
// ---------------------------------------------------------------------------
__global__ void head_kernel(const float* __restrict__ E, const int* __restrict__ S,
                            float* __restrict__ out) {
  const int bn = blockIdx.x;      // 0..15  (b*2 + n)
  const int d  = threadIdx.x;     // 0..255
  const int sid = S[bn * TT];     // S[b, n, t=0]
  out[bn * DIM + d] = E[sid * DIM + d];
  if (bn == 0 && d == 0) out[BB * NN * DIM] = 0.0f;   // loss accumulator
}

// ---------------------------------------------------------------------------
// Kernel 2: Esq[s] = ||E[s,:]||^2 into workspace. One wave per row.
// ---------------------------------------------------------------------------
__global__ void esq_kernel(const float* __restrict__ E, float* __restrict__ esq) {
  const int wave = (int)((blockIdx.x * blockDim.x + threadIdx.x) >> 5);
  const int lane = threadIdx.x & 31;
  if (wave >= SPK) return;
  const float* row = E + (size_t)wave * DIM;
  float s = 0.0f;
  #pragma unroll
  for (int d = lane; d < DIM; d += 32) { float x = row[d]; s = fmaf(x, x, s); }
  s = wave_reduce_add(s);
  if (lane == 0) esq[wave] = s;
}

// ---------------------------------------------------------------------------
// Kernel 3: Dmin permutation term. One wave per (b, t); 4000 waves total.
// loss += (alpha*Dmin - beta) / (B*T)
// ---------------------------------------------------------------------------
__global__ void dmin_kernel(const float* __restrict__ H, const int* __restrict__ S,
                            const float* __restrict__ E,
                            const float* __restrict__ alphap,
                            const float* __restrict__ betap,
                            float* __restrict__ loss) {
  const int w    = (int)((blockIdx.x * blockDim.x + threadIdx.x) >> 5); // 0..3999
  const int lane = threadIdx.x & 31;
  const int b = w / TT, t = w % TT;
  const float* H0 = H + (size_t)((b * NN + 0) * DIM) * TT + t;
  const float* H1 = H + (size_t)((b * NN + 1) * DIM) * TT + t;
  const int s0 = S[(b * NN + 0) * TT + t];
  const int s1 = S[(b * NN + 1) * TT + t];
  const float* E0 = E + (size_t)s0 * DIM;
  const float* E1 = E + (size_t)s1 * DIM;
  float p0 = 0.0f, p1 = 0.0f;
  #pragma unroll
  for (int d = lane; d < DIM; d += 32) {
    float h0 = H0[(size_t)d * TT], h1 = H1[(size_t)d * TT];
    float e0 = E0[d],              e1 = E1[d];
    float a0 = h0 - e0, a1 = h1 - e1;   // identity alignment
    float b0 = h0 - e1, b1 = h1 - e0;   // swapped alignment
    p0 = fmaf(a0, a0, fmaf(a1, a1, p0));
    p1 = fmaf(b0, b0, fmaf(b1, b1, p1));
  }
  p0 = wave_reduce_add(p0);
  p1 = wave_reduce_add(p1);
  if (lane == 0) {
    const float alpha = alphap[0], beta = betap[0];
    const float dmin = fminf(p0, p1);
    atomicAdd(loss, (alpha * dmin - beta) * (1.0f / (float)(BB * TT)));
  }
}

// ---------------------------------------------------------------------------
// Kernel 4: fused fp32 WMMA GEMM (E x H-columns) + streaming logsumexp over
// the 2000 speakers. One workgroup (8 waves) per 16-column tile.
// ---------------------------------------------------------------------------
__global__ void __launch_bounds__(256)
lse_kernel(const float* __restrict__ H, const float* __restrict__ E,
           const float* __restrict__ esq, const float* __restrict__ alphap,
           const float* __restrict__ betap, float* __restrict__ loss) {
  // Pair-interleaved B tile: element (k, col) at (k>>1)*32 + col*2 + (k&1)
  __shared__ float ldsB[DIM * 16];        // 16 KB
  __shared__ float hsq[16];
  __shared__ float wstate[8][16][2];      // per-wave (m, s) per column

  const int tid    = threadIdx.x;
  const int lane   = tid & 31;
  const int waveId = tid >> 5;
  const int l15    = lane & 15;
  const int half   = lane >> 4;           // 0 or 1
  const float alpha = alphap[0];
  const float beta  = betap[0];
  const float twoA  = 2.0f * alpha;

  if (tid < 16) hsq[tid] = 0.0f;

  // ---- stage 16 H columns into LDS via async-to-LDS (each lane scatters its
  //      own strided global element to its pair-interleaved LDS slot) ----
  const int col = tid & 15;
  const int d0  = tid >> 4;               // 0..15
  {
    const int c   = blockIdx.x * 16 + col;        // global column id
    const int b   = c / (NN * TT);
    const int rem = c % (NN * TT);
    const int n   = rem / TT;
    const int t   = rem % TT;
    const float* hp = H + (size_t)((b * NN + n) * DIM) * TT + t;
    #pragma unroll
    for (int i = 0; i < 16; ++i) {
      const int d = d0 + 16 * i;
      const unsigned loff =
          (unsigned)(size_t)&ldsB[(d >> 1) * 32 + col * 2 + (d & 1)];
      const unsigned long long gaddr =
          (unsigned long long)(size_t)(hp + (size_t)d * TT);
      asm volatile("global_load_async_to_lds_b32 %0, %1, off"
                   :: "v"(loff), "v"(gaddr) : "memory");
    }
    asm volatile("s_wait_asynccnt 0x0" ::: "memory");
  }
  __syncthreads();

  // ---- ||H_col||^2 from LDS (once per block) ----
  {
    float sq = 0.0f;
    #pragma unroll
    for (int i = 0; i < 16; ++i) {
      const int d = d0 + 16 * i;
      const float x = ldsB[(d >> 1) * 32 + col * 2 + (d & 1)];
      sq = fmaf(x, x, sq);
    }
    atomicAdd(&hsq[col], sq);
  }
  __syncthreads();

  // ---- stream s-tiles: WMMA fp32 GEMM + online logsumexp ----
  float runM = -3.0e38f;
  float runS = 0.0f;
  const float* bbase = ldsB + l15 * 2 + half * 32;

  for (int tile = waveId; tile < NSTILE; tile += 8) {
    const int srow0 = tile * 16;
    const float* aptr = E + (size_t)(srow0 + l15) * DIM + (half << 1);

    v8f acc = {0.f, 0.f, 0.f, 0.f, 0.f, 0.f, 0.f, 0.f};
    #pragma unroll 8
    for (int k = 0; k < DIM; k += 4) {
      v2f av = *(const v2f*)(aptr + k);                 // A: E rows (16x4)
      v2f bv = *(const v2f*)(bbase + (k >> 1) * 32);    // B: H cols (4x16)
      acc = __builtin_amdgcn_wmma_f32_16x16x4_f32(
          false, av, false, bv, (short)0, acc, false, false);
    }

    // epilogue: vals = 2a*cross - a*Esq[row]; rows of this lane = srow0+half*8+r
    const float* ep = esq + srow0 + half * 8;
    v4f e0 = *(const v4f*)(ep);
    v4f e1 = *(const v4f*)(ep + 4);

    float vv[8];
    float tmax = -3.0e38f;
    #pragma unroll
    for (int r = 0; r < 8; ++r) {
      const float esv = (r < 4) ? e0[r] : e1[r - 4];
      const float v = fmaf(twoA, acc[r], -alpha * esv);
      vv[r] = v;
      tmax = fmaxf(tmax, v);
    }
    tmax = fmaxf(tmax, swap16(tmax));                   // 16 rows per column
    float tsum = 0.0f;
    #pragma unroll
    for (int r = 0; r < 8; ++r) tsum += FEXP2((vv[r] - tmax) * LOG2E);
    tsum += swap16(tsum);

    // online merge of (tmax, tsum) into running state
    const float newM = fmaxf(runM, tmax);
    runS = runS * FEXP2((runM - newM) * LOG2E) + tsum * FEXP2((tmax - newM) * LOG2E);
    runM = newM;
  }

  if (lane < 16) { wstate[waveId][lane][0] = runM; wstate[waveId][lane][1] = runS; }
  __syncthreads();

  // ---- combine the 8 wave states per column, finish lse, atomic to loss ----
  if (tid < 16) {
    float M = -3.0e38f, Ssum = 0.0f;
    #pragma unroll
    for (int w = 0; w < 8; ++w) {
      const float m = wstate[w][tid][0];
      const float s = wstate[w][tid][1];
      const float nM = fmaxf(M, m);
      Ssum = Ssum * FEXP2((M - nM) * LOG2E) + s * FEXP2((m - nM) * LOG2E);
      M = nM;
    }
    const float lse = fmaf(-alpha, hsq[tid], beta) + M + FLOG2(Ssum) * LN2;
    atomicAdd(loss, lse * (1.0f / (float)NCOL));
  }
}

// ---------------------------------------------------------------------------
extern "C" void kernel_launch(void* const* d_in, const int* in_sizes, int n_in,
                              void* d_out, int out_size, void* d_ws, size_t ws_size,
                              hipStream_t stream) {
  const float* H     = (const float*)d_in[0];   // (8, 2, 256, 500)
  const int*   S     = (const int*)d_in[1];     // (8, 2, 500)
  const float* E     = (const float*)d_in[2];   // (2000, 256)
  const float* alpha = (const float*)d_in[3];   // (1,)
  const float* beta  = (const float*)d_in[4];   // (1,)
  float* out  = (float*)d_out;                  // [0..4095]=E_S head, [4096]=loss
  float* esq  = (float*)d_ws;                   // 2000 floats scratch
  float* lossp = out + BB * NN * DIM;

  head_kernel<<<16, 256, 0, stream>>>(E, S, out);          // also zeroes loss
  esq_kernel<<<250, 256, 0, stream>>>(E, esq);             // 2000 waves
  dmin_kernel<<<500, 256, 0, stream>>>(H, S, E, alpha, beta, lossp);
  lse_kernel<<<500, 256, 0, stream>>>(H, E, esq, alpha, beta, lossp);
}